// ConvAttn2D_14439680049186
// MI455X (gfx1250) — compile-verified
//
#include <hip/hip_runtime.h>
#include <hip/hip_bf16.h>

#define BB 8
#define CC 256
#define HH 128
#define WWID 128
#define HWP (HH*WWID)
#define AA 16
#define LKK 13
#define EPSF 1e-5f

typedef float v2f __attribute__((ext_vector_type(2)));
typedef float v8f __attribute__((ext_vector_type(8)));
typedef unsigned int u32x4 __attribute__((ext_vector_type(4)));
typedef int i32x4 __attribute__((ext_vector_type(4)));
typedef int i32x8 __attribute__((ext_vector_type(8)));

// ---------------- Kernel 1: global average pool over f_att = x[:, :16] ----------
__global__ __launch_bounds__(256) void k_pool(const float* __restrict__ x,
                                              float* __restrict__ g_raw) {
    int ba = blockIdx.x;                       // 0..B*A-1
    const float4* p = (const float4*)(x + (size_t)((ba >> 4) * CC + (ba & 15)) * HWP);
    float s = 0.f;
    for (int i = threadIdx.x; i < HWP / 4; i += 256) {
        float4 v = p[i];
        s += (v.x + v.y) + (v.z + v.w);
    }
    __shared__ float red[256];
    red[threadIdx.x] = s;
    __syncthreads();
    for (int off = 128; off > 0; off >>= 1) {
        if (threadIdx.x < off) red[threadIdx.x] += red[threadIdx.x + off];
        __syncthreads();
    }
    if (threadIdx.x == 0) g_raw[ba] = red[0] * (1.f / HWP);
}

// ---------------- Kernel 2: tiny MLP -> per-sample dynamic 3x3 kernels ----------
__global__ __launch_bounds__(256) void k_kgen(const float* __restrict__ g_raw,
                                              const float* __restrict__ w_kg1,
                                              const float* __restrict__ w_kg2,
                                              float* __restrict__ kern) {
    __shared__ float g2[BB * AA];
    int t = threadIdx.x;
    if (t < BB * AA) {
        int b = t >> 4, o = t & 15;
        float acc = 0.f;
        for (int i = 0; i < AA; ++i) acc = fmaf(w_kg1[o * AA + i], g_raw[b * AA + i], acc);
        g2[t] = acc > 0.f ? acc : 0.f;
    }
    __syncthreads();
    for (int idx = t; idx < BB * AA * 9; idx += 256) {
        int b = idx / 144, o = idx % 144;
        float acc = 0.f;
        for (int i = 0; i < AA; ++i) acc = fmaf(w_kg2[o * AA + i], g2[b * AA + i], acc);
        kern[idx] = acc;
    }
}

// ---------------- Kernel 3: depthwise 13x13 + dynamic 3x3 on f_att --------------
#define TS 32
#define THW (TS + LKK - 1)   // 44
__global__ __launch_bounds__(256) void k_conv(const float* __restrict__ x,
                                              const float* __restrict__ w_large,
                                              const float* __restrict__ kern,
                                              float* __restrict__ sbuf) {
    __shared__ float ft[THW * THW];
    __shared__ float wl[LKK * LKK];
    __shared__ float kd[9];
    int a = blockIdx.y, b = blockIdx.z;
    int tx = blockIdx.x & 3, ty = blockIdx.x >> 2;
    int h0 = ty * TS, w0 = tx * TS;
    int t = threadIdx.x;
    const float* fp = x + (size_t)(b * CC + a) * HWP;
    for (int idx = t; idx < THW * THW; idx += 256) {
        int iy = idx / THW, ix = idx % THW;
        int gh = h0 + iy - 6, gw = w0 + ix - 6;
        float v = 0.f;
        if (gh >= 0 && gh < HH && gw >= 0 && gw < WWID) v = fp[gh * WWID + gw];
        ft[idx] = v;
    }
    if (t < 169) wl[t] = w_large[a * 169 + t];
    if (t < 9)   kd[t] = kern[b * 144 + a * 9 + t];
    __syncthreads();

    // each thread: 4 consecutive output pixels in one row
    int q0 = 4 * t;
    int oy = q0 >> 5, ox = q0 & 31;
    float ax = 0.f, ay = 0.f, az = 0.f, aw = 0.f;
    for (int u = 0; u < LKK; ++u) {
        const float* row = &ft[(oy + u) * THW + ox];
        float r[16];
        #pragma unroll
        for (int k = 0; k < 16; ++k) r[k] = row[k];
        const float* wr = &wl[u * LKK];
        #pragma unroll
        for (int v = 0; v < LKK; ++v) {
            float w = wr[v];
            ax = fmaf(r[v],     w, ax);
            ay = fmaf(r[v + 1], w, ay);
            az = fmaf(r[v + 2], w, az);
            aw = fmaf(r[v + 3], w, aw);
        }
    }
    #pragma unroll
    for (int i = 0; i < 3; ++i) {
        const float* row = &ft[(oy + i + 5) * THW + ox + 5];
        float r[6];
        #pragma unroll
        for (int k = 0; k < 6; ++k) r[k] = row[k];
        #pragma unroll
        for (int j = 0; j < 3; ++j) {
            float w = kd[3 * i + j];
            ax = fmaf(r[j],     w, ax);
            ay = fmaf(r[j + 1], w, ay);
            az = fmaf(r[j + 2], w, az);
            aw = fmaf(r[j + 3], w, aw);
        }
    }
    float4 acc; acc.x = ax; acc.y = ay; acc.z = az; acc.w = aw;
    *(float4*)(sbuf + (size_t)(b * AA + a) * HWP + (h0 + oy) * WWID + (w0 + ox)) = acc;
}

// ---------------- Kernel 4: WMMA fusion GEMM + residual + LayerNorm -------------
#define NWV 2          // waves per block
#define WFS_RS 17      // w_fusion LDS row stride (scalar reads)
#define OT_RS 20       // out-tile LDS row stride: 80B = 5*16B -> B128-aligned,
                       // and 20*lo mod 64 is a 16-cycle permutation -> no conflicts
__global__ __launch_bounds__(NWV * 32) void k_fuse_ln(const float* __restrict__ x,
                                                      const float* __restrict__ sbuf,
                                                      const float* __restrict__ wf,
                                                      const float* __restrict__ gamma,
                                                      const float* __restrict__ beta,
                                                      float* __restrict__ out) {
    __shared__ float wfs[CC * WFS_RS];
    __shared__ float gam[CC], bet[CC];
    __shared__ __align__(16) float ot[NWV][CC * OT_RS];
    __shared__ float muL[NWV][16], rsL[NWV][16];
    int t = threadIdx.x;
    int wave = __builtin_amdgcn_readfirstlane(t >> 5);   // wave-uniform in SGPR
    int lane = t & 31;
    int lo = lane & 15, hi = lane >> 4;
    int c8 = lane >> 2;            // 0..7  (streaming phases)
    int p4 = (lane & 3) * 4;       // 0,4,8,12
    int b  = blockIdx.x >> 9;                        // 512 blocks per batch
    int p0 = ((blockIdx.x & 511) * (NWV * 16)) + wave * 16;

    // preload w_fusion / gamma / beta into LDS (whole block)
    for (int idx = t; idx < CC * AA; idx += NWV * 32)
        wfs[(idx >> 4) * WFS_RS + (idx & 15)] = wf[idx];
    for (int idx = t; idx < CC; idx += NWV * 32) { gam[idx] = gamma[idx]; bet[idx] = beta[idx]; }

    // ---- stream x tile (residual) into this wave's LDS tile --------------------
    // TDM: one 2D descriptor per wave: 256 rows x 16 floats, row stride 16384
    // floats, LDS padding 4 dwords after every 16 dwords -> OT_RS = 20 layout.
    const float* xb = x + (size_t)b * CC * HWP + p0;
    float* o = ot[wave];
#if __has_builtin(__builtin_amdgcn_tensor_load_to_lds) && defined(__gfx1250__)
    {
        typedef __attribute__((address_space(3))) float lds_float_t;
        unsigned lds_base = (unsigned)(unsigned long long)(lds_float_t*)o;
        unsigned long long ga = (unsigned long long)xb;
        u32x4 g0;
        g0[0] = 1u;                                             // count=1 (valid), user mode
        g0[1] = lds_base;                                       // lds_addr
        g0[2] = (unsigned)(ga & 0xffffffffull);                 // global_addr[31:0]
        g0[3] = (unsigned)((ga >> 32) & 0x1ffffffull)           // global_addr[56:32]
              | (2u << 30);                                     // type = 2 (image)
        i32x8 g1;
        g1[0] = (int)((2u << 16)        // data_size = 4B
                    | (1u << 20)        // pad_enable
                    | (3u << 22)        // pad_interval: 16 dwords
                    | (3u << 25));      // pad_amount:   4 dwords
        g1[1] = (int)((16384u & 0xffffu) << 16);                // tensor_dim0[15:0]
        g1[2] = (int)((16384u >> 16) | (256u << 16));           // dim0[31:16] | tensor_dim1[15:0]
        g1[3] = (int)((256u >> 16) | (16u << 16));              // dim1[31:16] | tile_dim0 = 16
        g1[4] = (int)(256u);                                    // tile_dim1 = 256, tile_dim2 = 0
        g1[5] = (int)16384;                                     // tensor_dim0_stride[31:0]
        g1[6] = 0;                                              // stride0[47:32] | stride1[15:0]
        g1[7] = 0;
        i32x4 zz = {0, 0, 0, 0};
#if __clang_major__ >= 23
        i32x8 z8 = {};
        __builtin_amdgcn_tensor_load_to_lds(g0, g1, zz, zz, z8, 0);
#else
        __builtin_amdgcn_tensor_load_to_lds(g0, g1, zz, zz, 0);
#endif
    }
    __builtin_amdgcn_s_wait_tensorcnt((short)0);
#else
    for (int it = 0; it < 32; ++it) {
        int c = it * 8 + c8;
        float4 v = *(const float4*)(xb + (size_t)c * HWP + p4);
        *(float4*)&o[c * OT_RS + p4] = v;
    }
#endif
    __syncthreads();

    // A fragments: 16 pixels x 16 attn channels of (f_dyn + f_large)
    // ISA 16x4 f32 layout: lane = lo(+16*hi), VGPR v holds K = 4*kk + v + 2*hi
    v2f af[4];
    const float* sb = sbuf + (size_t)b * AA * HWP + p0 + lo;
    #pragma unroll
    for (int kk = 0; kk < 4; ++kk)
        #pragma unroll
        for (int v = 0; v < 2; ++v)
            af[kk][v] = sb[(size_t)(4 * kk + v + 2 * hi) * HWP];

    // 16 channel tiles of 16; K=16 via 4 chained 16x16x4 f32 WMMAs
    for (int ct = 0; ct < 16; ++ct) {
        v8f acc = {};
        #pragma unroll
        for (int kk = 0; kk < 4; ++kk) {
            v2f bm;
            bm[0] = wfs[(ct * 16 + lo) * WFS_RS + 4 * kk + 2 * hi];
            bm[1] = wfs[(ct * 16 + lo) * WFS_RS + 4 * kk + 2 * hi + 1];
            acc = __builtin_amdgcn_wmma_f32_16x16x4_f32(
                false, af[kk], false, bm, (short)0, acc, false, false);
        }
        // D layout: VGPR r -> pixel r + 8*hi, N -> channel ct*16 + lo
        #pragma unroll
        for (int r = 0; r < 8; ++r)
            o[(ct * 16 + lo) * OT_RS + r + 8 * hi] += acc[r];
    }
    __syncthreads();

    // LayerNorm stats per pixel (= lo); each half-wave sums 128 channels
    float sum = 0.f, sq = 0.f;
    for (int i = 0; i < 128; ++i) {
        float v = o[(hi * 128 + i) * OT_RS + lo];
        sum += v;
        sq = fmaf(v, v, sq);
    }
    sum += __shfl_xor(sum, 16, 32);
    sq  += __shfl_xor(sq, 16, 32);
    float mu  = sum * (1.f / CC);
    float var = sq * (1.f / CC) - mu * mu;
    float rsg = rsqrtf(var + EPSF);
    if (hi == 0) { muL[wave][lo] = mu; rsL[wave][lo] = rsg; }
    __syncthreads();

    // normalize + affine, vectorized float4 stores (lane owns 4 pixels)
    float mu4[4], rs4[4];
    #pragma unroll
    for (int j = 0; j < 4; ++j) { mu4[j] = muL[wave][p4 + j]; rs4[j] = rsL[wave][p4 + j]; }
    float* ob = out + (size_t)b * CC * HWP + p0;
    for (int it = 0; it < 32; ++it) {
        int c = it * 8 + c8;
        float4 v = *(const float4*)&o[c * OT_RS + p4];
        float g = gam[c], be = bet[c];
        float4 r;
        r.x = fmaf((v.x - mu4[0]) * rs4[0], g, be);
        r.y = fmaf((v.y - mu4[1]) * rs4[1], g, be);
        r.z = fmaf((v.z - mu4[2]) * rs4[2], g, be);
        r.w = fmaf((v.w - mu4[3]) * rs4[3], g, be);
        *(float4*)(ob + (size_t)c * HWP + p4) = r;
    }
}

extern "C" void kernel_launch(void* const* d_in, const int* in_sizes, int n_in,
                              void* d_out, int out_size, void* d_ws, size_t ws_size,
                              hipStream_t stream) {
    const float* x        = (const float*)d_in[0];
    const float* w_kg1    = (const float*)d_in[1];
    const float* w_kg2    = (const float*)d_in[2];
    const float* w_large  = (const float*)d_in[3];
    const float* w_fusion = (const float*)d_in[4];
    const float* ln_gamma = (const float*)d_in[5];
    const float* ln_beta  = (const float*)d_in[6];
    float* out = (float*)d_out;

    float* ws    = (float*)d_ws;
    float* g_raw = ws;            // 128 floats
    float* kern  = ws + 128;      // 1152 floats
    float* sbuf  = ws + 2048;     // B*A*H*W = 2,097,152 floats (8 MiB)

    k_pool<<<BB * AA, 256, 0, stream>>>(x, g_raw);
    k_kgen<<<1, 256, 0, stream>>>(g_raw, w_kg1, w_kg2, kern);
    k_conv<<<dim3(16, AA, BB), 256, 0, stream>>>(x, w_large, kern, sbuf);
    k_fuse_ln<<<BB * 512, NWV * 32, 0, stream>>>(x, sbuf, w_fusion, ln_gamma, ln_beta, out);
}